// DigitCaps_29540785062237
// MI455X (gfx1250) — compile-verified
//
#include <hip/hip_runtime.h>
#include <math.h>

// Problem constants (from reference)
constexpr int Bsz = 128;   // batch
constexpr int R   = 1152;  // routes
constexpr int C   = 10;    // capsules
constexpr int O   = 16;    // out channels
constexpr int I   = 338;   // in channels
constexpr int CO  = C * O; // 160

// GEMM tiling
constexpr int KSLAB = 40;          // K-slab per LDS stage
constexpr int KPAD  = 42;          // padded LDS row stride (bank spread)
constexpr int NSLABS = (I + KSLAB - 1) / KSLAB; // 9 (last slab zero-padded)

typedef float v2f __attribute__((ext_vector_type(2)));
typedef float v8f __attribute__((ext_vector_type(8)));

// ---------------------------------------------------------------------------
// Kernel 1: u_hat[r][co][b] = sum_i W[r,co,i] * x[b,r,i]
// One block per route r. 256 threads = 8 waves; wave w owns N-tile (batches
// w*16..w*16+15) and accumulates all 10 M-tiles in registers (10 x v8f).
// fp32 WMMA 16x16x4, K staged through LDS in slabs of 40.
// ---------------------------------------------------------------------------
__global__ __launch_bounds__(256)
void gemm_uhat_kernel(const float* __restrict__ x,   // [B][R][I]
                      const float* __restrict__ W,   // [R][C][O][I]
                      float* __restrict__ u_hat)     // [R][CO][B]
{
    __shared__ float a_s[CO  * KPAD];  // W slab: 160 x 42 floats (26.9 KB)
    __shared__ float x_s[Bsz * KPAD];  // x slab: 128 x 42 floats (21.5 KB)

    const int r    = blockIdx.x;
    const int tid  = threadIdx.x;
    const int wave = tid >> 5;        // 0..7 -> N-tile
    const int lane = tid & 31;
    const int n0   = wave * 16;

    const int mrow = lane & 15;        // row within 16 (M for A-frag, N for B-frag)
    const int koff = (lane >> 4) * 2;  // upper half-wave handles K+2

    v8f acc[10] = {};                  // 10 M-tiles x (16x16 f32 tile / wave)

    for (int s = 0; s < NSLABS; ++s) {
        const int k0 = s * KSLAB;
        __syncthreads();               // protect LDS reuse across slabs

        // Stage x slab: x_s[b][kk] = x[b, r, k0+kk]  (zero-pad k >= I)
        for (int idx = tid; idx < Bsz * KSLAB; idx += 256) {
            const int b  = idx / KSLAB;
            const int kk = idx - b * KSLAB;
            const int k  = k0 + kk;
            x_s[b * KPAD + kk] = (k < I) ? x[(size_t)(b * R + r) * I + k] : 0.0f;
        }
        // Stage W slab: a_s[m][kk] = W[r, m, k0+kk]
        for (int idx = tid; idx < CO * KSLAB; idx += 256) {
            const int m  = idx / KSLAB;
            const int kk = idx - m * KSLAB;
            const int k  = k0 + kk;
            a_s[m * KPAD + kk] = (k < I) ? W[(size_t)(r * CO + m) * I + k] : 0.0f;
        }
        __syncthreads();

        for (int kq = 0; kq < KSLAB; kq += 4) {
            // B fragment (4x16, K x N): VGPR j, lane l -> (K=kq+koff+j, N=l&15)
            const float* bp = &x_s[(n0 + mrow) * KPAD + kq + koff];
            v2f bf;
            bf.x = bp[0];
            bf.y = bp[1];
#pragma unroll
            for (int m = 0; m < 10; ++m) {
                // A fragment (16x4, M x K): lane l -> M=l&15, VGPRs K=kq+koff+{0,1}
                const float* ap = &a_s[(m * 16 + mrow) * KPAD + kq + koff];
                v2f af;
                af.x = ap[0];
                af.y = ap[1];
                acc[m] = __builtin_amdgcn_wmma_f32_16x16x4_f32(
                    false, af, false, bf, (short)0, acc[m], false, false);
            }
        }
    }

    // Store D tiles: VGPR v, lanes 0-15 -> M=v, lanes 16-31 -> M=v+8; N=lane&15
    const int col = n0 + mrow;
    const int mhi = (lane >> 4) * 8;
#pragma unroll
    for (int m = 0; m < 10; ++m) {
#pragma unroll
        for (int v = 0; v < 8; ++v) {
            const int row = m * 16 + v + mhi;
            u_hat[((size_t)r * CO + row) * Bsz + col] = acc[m][v];
        }
    }
}

// ---------------------------------------------------------------------------
// Zero b_ij logits
// ---------------------------------------------------------------------------
__global__ void zero_kernel(float* __restrict__ p, int n)
{
    const int i = blockIdx.x * blockDim.x + threadIdx.x;
    if (i < n) p[i] = 0.0f;
}

// ---------------------------------------------------------------------------
// Softmax over routes axis: cij[r][c] = exp(b[r][c]-max_r)/sum_r exp(...)
// One block per capsule column c (10 blocks x 128 threads).
// ---------------------------------------------------------------------------
__global__ __launch_bounds__(128)
void softmax_routes_kernel(const float* __restrict__ b_ij,  // [R][C]
                           float* __restrict__ cij)         // [R][C]
{
    __shared__ float red[128];
    const int c = blockIdx.x;
    const int t = threadIdx.x;

    float mx = -INFINITY;
    for (int r = t; r < R; r += 128) mx = fmaxf(mx, b_ij[r * C + c]);
    red[t] = mx;
    __syncthreads();
    for (int off = 64; off > 0; off >>= 1) {
        if (t < off) red[t] = fmaxf(red[t], red[t + off]);
        __syncthreads();
    }
    mx = red[0];
    __syncthreads();

    float sum = 0.0f;
    for (int r = t; r < R; r += 128) sum += expf(b_ij[r * C + c] - mx);
    red[t] = sum;
    __syncthreads();
    for (int off = 64; off > 0; off >>= 1) {
        if (t < off) red[t] += red[t + off];
        __syncthreads();
    }
    const float inv = 1.0f / red[0];

    for (int r = t; r < R; r += 128)
        cij[r * C + c] = expf(b_ij[r * C + c] - mx) * inv;
}

// ---------------------------------------------------------------------------
// s_partial[chunk][co][b] = sum_{r in chunk} cij[r][c] * u_hat[r][co][b]
// grid (160, 9), block 128 (thread = batch b). Coalesced 512B rows.
// ---------------------------------------------------------------------------
constexpr int RCHUNKS = 9;
constexpr int RPERCHUNK = R / RCHUNKS; // 128

__global__ __launch_bounds__(128)
void s_reduce_kernel(const float* __restrict__ u_hat,   // [R][CO][B]
                     const float* __restrict__ cij,     // [R][C]
                     float* __restrict__ s_partial)     // [RCHUNKS][CO][B]
{
    const int co    = blockIdx.x;
    const int chunk = blockIdx.y;
    const int b     = threadIdx.x;
    const int c     = co >> 4;

    float acc = 0.0f;
    const int r0 = chunk * RPERCHUNK;
#pragma unroll 4
    for (int rr = 0; rr < RPERCHUNK; ++rr) {
        const int r = r0 + rr;
        acc += cij[r * C + c] * u_hat[((size_t)r * CO + co) * Bsz + b];
    }
    s_partial[(chunk * CO + co) * Bsz + b] = acc;
}

// ---------------------------------------------------------------------------
// v[b][c][o] = squash(s) = s*|s| / (1+s^2)   (elementwise, per reference)
// ---------------------------------------------------------------------------
__global__ __launch_bounds__(128)
void squash_kernel(const float* __restrict__ s_partial, // [RCHUNKS][CO][B]
                   float* __restrict__ vout)            // [B][C][O]
{
    const int co = blockIdx.x;
    const int b  = threadIdx.x;

    float s = 0.0f;
#pragma unroll
    for (int ch = 0; ch < RCHUNKS; ++ch)
        s += s_partial[(ch * CO + co) * Bsz + b];

    const float sq = s * s;
    const float v  = s * fabsf(s) / (1.0f + sq);

    const int c = co >> 4;
    const int o = co & 15;
    vout[(b * C + c) * O + o] = v;
}

// ---------------------------------------------------------------------------
// Agreement: b_ij[r][c] += (1/B) * sum_{b,o} u_hat[r][c*16+o][b] * v[b][c][o]
// One block per route (1152 blocks x 320 threads = 10 waves, wave = capsule c).
// ---------------------------------------------------------------------------
__global__ __launch_bounds__(320)
void agreement_kernel(const float* __restrict__ u_hat,  // [R][CO][B]
                      const float* __restrict__ v,      // [B][C][O]
                      float* __restrict__ b_ij)         // [R][C]
{
    const int r    = blockIdx.x;
    const int c    = threadIdx.x >> 5;   // 0..9
    const int lane = threadIdx.x & 31;

    float acc = 0.0f;
    // 2048 = O(16) x B(128) elements per (r,c); lanes stride 32 -> coalesced b
    for (int j = lane; j < O * Bsz; j += 32) {
        const int o = j >> 7;       // j / 128
        const int b = j & 127;      // j % 128
        acc += u_hat[((size_t)r * CO + (c * O + o)) * Bsz + b] *
               v[(b * C + c) * O + o];
    }
#pragma unroll
    for (int off = 16; off > 0; off >>= 1)
        acc += __shfl_down(acc, off, 32);

    if (lane == 0)
        b_ij[r * C + c] += acc * (1.0f / (float)Bsz);
}

// ---------------------------------------------------------------------------
// Host launch
// ---------------------------------------------------------------------------
extern "C" void kernel_launch(void* const* d_in, const int* in_sizes, int n_in,
                              void* d_out, int out_size, void* d_ws, size_t ws_size,
                              hipStream_t stream) {
    (void)in_sizes; (void)n_in; (void)out_size; (void)ws_size;

    const float* x = (const float*)d_in[0];   // [B][R][I]
    const float* W = (const float*)d_in[1];   // [R][C][O][I]
    float* out = (float*)d_out;               // [B][C][O][1]

    // Workspace layout (floats)
    float* ws        = (float*)d_ws;
    float* u_hat     = ws;                                   // 23,592,960
    float* s_partial = u_hat + (size_t)R * CO * Bsz;         //    184,320
    float* v_ws      = s_partial + (size_t)RCHUNKS * CO * Bsz; //    20,480
    float* b_ij      = v_ws + (size_t)Bsz * C * O;           //     11,520
    float* cij       = b_ij + (size_t)R * C;                 //     11,520

    // b_ij = 0
    zero_kernel<<<(R * C + 255) / 256, 256, 0, stream>>>(b_ij, R * C);

    // u_hat GEMM (fp32 WMMA)
    gemm_uhat_kernel<<<R, 256, 0, stream>>>(x, W, u_hat);

    // 3 routing iterations
    for (int it = 0; it < 3; ++it) {
        softmax_routes_kernel<<<C, 128, 0, stream>>>(b_ij, cij);
        s_reduce_kernel<<<dim3(CO, RCHUNKS), 128, 0, stream>>>(u_hat, cij, s_partial);
        float* vdst = (it == 2) ? out : v_ws;
        squash_kernel<<<CO, 128, 0, stream>>>(s_partial, vdst);
        if (it < 2)
            agreement_kernel<<<R, 320, 0, stream>>>(u_hat, v_ws, b_ij);
    }
}